// EmbedMatcher_68040871903505
// MI455X (gfx1250) — compile-verified
//
#include <hip/hip_runtime.h>
#include <math.h>

// ---------------------------------------------------------------------------
// EmbedMatcher forward for MI455X (gfx1250, wave32, WMMA f32 16x16x4).
// Algebraic hoists vs reference:
//   * sum_k(concat @ W^T) == (sum_k concat) @ W^T + K*b   (13.4 GF -> 67 MF)
//   * q @ w_ih^T + b_ih + b_hh precomputed once (step-invariant)
// This revision register-blocks the GEMMs over N (8 accumulator tiles per
// wave) so each A fragment load is amortized over 8 WMMAs: fragment loads
// per v_wmma drop from 2.0 to ~1.13 in the hot loops.
// ---------------------------------------------------------------------------

typedef __attribute__((ext_vector_type(2))) float v2f;
typedef __attribute__((ext_vector_type(8))) float v8f;

#define EM_D   128
#define EM_2D  256
#define EM_B   1024
#define EM_K   200
#define EM_8D  1024

static __device__ __forceinline__ v8f wmma4(v2f a, v2f b, v8f c) {
  // D = A(16x4, f32) * B(4x16, f32) + C(16x16, f32)
  return __builtin_amdgcn_wmma_f32_16x16x4_f32(
      /*neg_a=*/false, a, /*neg_b=*/false, b,
      /*c_mod=*/(short)0, c, /*reuse_a=*/false, /*reuse_b=*/false);
}

// A fragment: X row-major (ld), rows m0..m0+15, K-slice k0..k0+3.
// Lanes 0-15: {K=k0, k0+1} for row=lane; lanes 16-31: {K=k0+2, k0+3}.
static __device__ __forceinline__ v2f load_a(const float* X, int ld, int m0, int k0) {
  int lane = threadIdx.x & 31;
  int r    = lane & 15;
  int kk   = (lane >> 4) << 1;
  const float* p = X + (m0 + r) * ld + k0 + kk;
  v2f a; a.x = p[0]; a.y = p[1];
  return a;
}

// B fragment for X @ W^T: B[k][n] = W[n][k], W row-major (N_out, K_in).
static __device__ __forceinline__ v2f load_bT(const float* W, int ld, int n0, int k0) {
  int lane = threadIdx.x & 31;
  int n    = lane & 15;
  int kk   = (lane >> 4) << 1;
  const float* p = W + (n0 + n) * ld + k0 + kk;
  v2f b; b.x = p[0]; b.y = p[1];
  return b;
}

// B fragment, row-major Bm (K x N): B[k][n].
static __device__ __forceinline__ v2f load_b(const float* Bm, int ld, int k0, int n0) {
  int lane = threadIdx.x & 31;
  int n    = lane & 15;
  int kk   = (lane >> 4) << 1;
  const float* p = Bm + (k0 + kk) * ld + n0 + n;
  v2f b; b.x = p[0]; b.y = p[ld];
  return b;
}

static __device__ __forceinline__ float sigf(float x) { return 1.0f / (1.0f + expf(-x)); }

#define V8ZERO {0.f,0.f,0.f,0.f,0.f,0.f,0.f,0.f}

// ---------------------------------------------------------------------------
// K1: S[b][0:128] = sum_k emb[rel[b,k]], S[b][128:256] = sum_k emb[ent[b,k]]
// One block per batch row, 8 waves, float4 lanes, LDS cross-wave reduce.
// ---------------------------------------------------------------------------
__global__ void k_gather_sum(const int* __restrict__ rel, const int* __restrict__ ent,
                             const float* __restrict__ emb, float* __restrict__ S) {
  __shared__ float sp[8][EM_2D];
  const int b = blockIdx.x;
  const int w = threadIdx.x >> 5, lane = threadIdx.x & 31;
  const float4* embv = (const float4*)emb;
  float4 ar = {0.f,0.f,0.f,0.f}, ae = {0.f,0.f,0.f,0.f};
  for (int k = w; k < EM_K; k += 8) {
    int ir = rel[b * EM_K + k];
    int ie = ent[b * EM_K + k];
    float4 vr = embv[(long)ir * 32 + lane];
    float4 ve = embv[(long)ie * 32 + lane];
    ar.x += vr.x; ar.y += vr.y; ar.z += vr.z; ar.w += vr.w;
    ae.x += ve.x; ae.y += ve.y; ae.z += ve.z; ae.w += ve.w;
  }
  sp[w][lane * 4 + 0] = ar.x; sp[w][lane * 4 + 1] = ar.y;
  sp[w][lane * 4 + 2] = ar.z; sp[w][lane * 4 + 3] = ar.w;
  sp[w][EM_D + lane * 4 + 0] = ae.x; sp[w][EM_D + lane * 4 + 1] = ae.y;
  sp[w][EM_D + lane * 4 + 2] = ae.z; sp[w][EM_D + lane * 4 + 3] = ae.w;
  __syncthreads();
  int c = threadIdx.x;  // 0..255
  float s = 0.f;
  #pragma unroll
  for (int ww = 0; ww < 8; ++ww) s += sp[ww][c];
  S[(long)b * EM_2D + c] = s;
}

// ---------------------------------------------------------------------------
// K2: support = tanh((S @ gcn_w^T + 200*b)/1024); FFN + residual + LayerNorm.
// Grid: 64 row-tiles of 16; block 128 (4 waves). N-blocked WMMA (A reused).
// ---------------------------------------------------------------------------
__global__ void k_support(const float* __restrict__ S,
                          const float* __restrict__ gcn_w, const float* __restrict__ gcn_b,
                          const float* __restrict__ p1w, const float* __restrict__ p1b,
                          const float* __restrict__ p2w, const float* __restrict__ p2b,
                          const float* __restrict__ ln_a, const float* __restrict__ ln_b,
                          float* __restrict__ sg) {
  __shared__ float s_sup[16][EM_D];
  __shared__ float s_hid[16][EM_2D];
  __shared__ float s_out[16][EM_D];
  const int m0 = blockIdx.x * 16;
  const int w = threadIdx.x >> 5, lane = threadIdx.x & 31;
  const int col = lane & 15, rofs = (lane >> 4) << 3;

  // support = tanh((S @ gcn_w^T + 200*b) / 1024)   [N=128 -> 2 tiles/wave]
  {
    int n0 = w * 32;
    v8f acc[2] = {V8ZERO, V8ZERO};
    for (int k0 = 0; k0 < EM_2D; k0 += 4) {
      v2f a = load_a(S, EM_2D, m0, k0);
      #pragma unroll
      for (int t = 0; t < 2; ++t)
        acc[t] = wmma4(a, load_bT(gcn_w, EM_2D, n0 + t * 16, k0), acc[t]);
    }
    #pragma unroll
    for (int t = 0; t < 2; ++t) {
      float bias = gcn_b[n0 + t * 16 + col] * (float)EM_K;
      #pragma unroll
      for (int r = 0; r < 8; ++r)
        s_sup[rofs + r][n0 + t * 16 + col] = tanhf((acc[t][r] + bias) * (1.0f / 1024.0f));
    }
  }
  __syncthreads();

  // hidden = relu(support @ p1w^T + p1b)   [N=256 -> 4 tiles/wave]
  {
    int n0 = w * 64;
    v8f acc[4] = {V8ZERO, V8ZERO, V8ZERO, V8ZERO};
    for (int k0 = 0; k0 < EM_D; k0 += 4) {
      v2f a = load_a(&s_sup[0][0], EM_D, 0, k0);
      #pragma unroll
      for (int t = 0; t < 4; ++t)
        acc[t] = wmma4(a, load_bT(p1w, EM_D, n0 + t * 16, k0), acc[t]);
    }
    #pragma unroll
    for (int t = 0; t < 4; ++t) {
      float bias = p1b[n0 + t * 16 + col];
      #pragma unroll
      for (int r = 0; r < 8; ++r) {
        float v = acc[t][r] + bias;
        s_hid[rofs + r][n0 + t * 16 + col] = v > 0.f ? v : 0.f;
      }
    }
  }
  __syncthreads();

  // out2 = hidden @ p2w^T + p2b + support   [N=128 -> 2 tiles/wave]
  {
    int n0 = w * 32;
    v8f acc[2] = {V8ZERO, V8ZERO};
    for (int k0 = 0; k0 < EM_2D; k0 += 4) {
      v2f a = load_a(&s_hid[0][0], EM_2D, 0, k0);
      #pragma unroll
      for (int t = 0; t < 2; ++t)
        acc[t] = wmma4(a, load_bT(p2w, EM_2D, n0 + t * 16, k0), acc[t]);
    }
    #pragma unroll
    for (int t = 0; t < 2; ++t) {
      float bias = p2b[n0 + t * 16 + col];
      #pragma unroll
      for (int r = 0; r < 8; ++r)
        s_out[rofs + r][n0 + t * 16 + col] =
            acc[t][r] + bias + s_sup[rofs + r][n0 + t * 16 + col];
    }
  }
  __syncthreads();

  // LayerNorm (unbiased std, ddof=1): 4 rows per wave, wave32 shuffle reduce.
  for (int i = 0; i < 4; ++i) {
    int row = w * 4 + i;
    int c4 = lane * 4;
    float x0 = s_out[row][c4 + 0], x1 = s_out[row][c4 + 1];
    float x2 = s_out[row][c4 + 2], x3 = s_out[row][c4 + 3];
    float s = x0 + x1 + x2 + x3;
    for (int off = 16; off > 0; off >>= 1) s += __shfl_xor(s, off, 32);
    float mu = s * (1.0f / 128.0f);
    float d0 = x0 - mu, d1 = x1 - mu, d2 = x2 - mu, d3 = x3 - mu;
    float q = d0 * d0 + d1 * d1 + d2 * d2 + d3 * d3;
    for (int off = 16; off > 0; off >>= 1) q += __shfl_xor(q, off, 32);
    float inv = 1.0f / (sqrtf(q * (1.0f / 127.0f)) + 1e-3f);
    long o = (long)(m0 + row) * EM_D + c4;
    sg[o + 0] = d0 * inv * ln_a[c4 + 0] + ln_b[c4 + 0];
    sg[o + 1] = d1 * inv * ln_a[c4 + 1] + ln_b[c4 + 1];
    sg[o + 2] = d2 * inv * ln_a[c4 + 2] + ln_b[c4 + 2];
    sg[o + 3] = d3 * inv * ln_a[c4 + 3] + ln_b[c4 + 3];
  }
}

// ---------------------------------------------------------------------------
// K3a: q_emb gather (one block per row).
// ---------------------------------------------------------------------------
__global__ void k_qemb(const int* __restrict__ query, const float* __restrict__ emb,
                       float* __restrict__ q_emb) {
  int row = blockIdx.x, lane = threadIdx.x;
  int idx = query[row];
  ((float4*)q_emb)[(long)row * 32 + lane] = ((const float4*)emb)[(long)idx * 32 + lane];
}

// ---------------------------------------------------------------------------
// K3b: qW = emb[query] @ w_ih^T + b_ih + b_hh   (step-invariant, done once).
// Gather folded into A loads. Grid 64, block 256; 8 N-tiles register-blocked
// per wave: one A fragment load feeds 8 WMMAs.
// ---------------------------------------------------------------------------
__global__ void k_qw(const int* __restrict__ query, const float* __restrict__ emb,
                     const float* __restrict__ w_ih,
                     const float* __restrict__ b_ih, const float* __restrict__ b_hh,
                     float* __restrict__ qW) {
  const int m0 = blockIdx.x * 16;
  const int w = threadIdx.x >> 5, lane = threadIdx.x & 31;
  const int col = lane & 15, rofs = (lane >> 4) << 3;
  const int khalf = (lane >> 4) << 1;
  const float* arow = emb + (long)query[m0 + (lane & 15)] * EM_D + khalf;
  const int nbase = w * 128;
  v8f acc[8] = {V8ZERO, V8ZERO, V8ZERO, V8ZERO, V8ZERO, V8ZERO, V8ZERO, V8ZERO};
  for (int k0 = 0; k0 < EM_D; k0 += 4) {
    v2f a; a.x = arow[k0]; a.y = arow[k0 + 1];
    #pragma unroll
    for (int t = 0; t < 8; ++t)
      acc[t] = wmma4(a, load_bT(w_ih, EM_D, nbase + t * 16, k0), acc[t]);
  }
  #pragma unroll
  for (int t = 0; t < 8; ++t) {
    int n0 = nbase + t * 16;
    float bias = b_ih[n0 + col] + b_hh[n0 + col];
    #pragma unroll
    for (int r = 0; r < 8; ++r)
      qW[(long)(m0 + rofs + r) * EM_8D + n0 + col] = acc[t][r] + bias;
  }
}

// ---------------------------------------------------------------------------
// K4: zero h_r and c (d_ws is poisoned, reference starts LSTM state at 0).
// ---------------------------------------------------------------------------
__global__ void k_zero2(float* __restrict__ h_r, float* __restrict__ c) {
  int i = blockIdx.x * blockDim.x + threadIdx.x;
  h_r[i] = 0.f; c[i] = 0.f;
}

// ---------------------------------------------------------------------------
// K5: gates = qW + h_r @ w_hh^T   (B x 1024, K=256). 8 N-tiles per wave.
// ---------------------------------------------------------------------------
__global__ void k_gates(const float* __restrict__ h_r, const float* __restrict__ w_hh,
                        const float* __restrict__ qW, float* __restrict__ gates) {
  const int m0 = blockIdx.x * 16;
  const int w = threadIdx.x >> 5, lane = threadIdx.x & 31;
  const int col = lane & 15, rofs = (lane >> 4) << 3;
  const int nbase = w * 128;
  v8f acc[8] = {V8ZERO, V8ZERO, V8ZERO, V8ZERO, V8ZERO, V8ZERO, V8ZERO, V8ZERO};
  for (int k0 = 0; k0 < EM_2D; k0 += 4) {
    v2f a = load_a(h_r, EM_2D, m0, k0);
    #pragma unroll
    for (int t = 0; t < 8; ++t)
      acc[t] = wmma4(a, load_bT(w_hh, EM_2D, nbase + t * 16, k0), acc[t]);
  }
  #pragma unroll
  for (int t = 0; t < 8; ++t) {
    #pragma unroll
    for (int r = 0; r < 8; ++r) {
      long o = (long)(m0 + rofs + r) * EM_8D + nbase + t * 16 + col;
      gates[o] = acc[t][r] + qW[o];
    }
  }
}

// ---------------------------------------------------------------------------
// K6: LSTM elementwise: c' = sig(f)*c + sig(i)*tanh(g); h_out = q + sig(o)*tanh(c')
// writes h_out into h_r[:, :128].
// ---------------------------------------------------------------------------
__global__ void k_lstm(const float* __restrict__ gates, const float* __restrict__ q_emb,
                       float* __restrict__ c, float* __restrict__ h_r) {
  int idx = blockIdx.x * blockDim.x + threadIdx.x;  // over B*256
  int row = idx >> 8, col = idx & 255;
  const float* g = gates + (long)row * EM_8D;
  float gi = g[col], gf = g[EM_2D + col], gg = g[2 * EM_2D + col], go = g[3 * EM_2D + col];
  float cn = sigf(gf) * c[idx] + sigf(gi) * tanhf(gg);
  c[idx] = cn;
  if (col < EM_D)
    h_r[(long)row * EM_2D + col] = q_emb[(long)row * EM_D + col] + sigf(go) * tanhf(cn);
}

// ---------------------------------------------------------------------------
// K7: attention: logits = h_out @ sg^T -> softmax(rows) -> r = attn @ sg,
// writes r into h_r[:, 128:]. 16 rows/block, 8 waves, 64 KB LDS logits tile.
// Logits phase is 8-way N-blocked (A fragment reused across 8 WMMAs).
// ---------------------------------------------------------------------------
__global__ void k_attn(float* __restrict__ h_r, const float* __restrict__ sg) {
  __shared__ float s_l[16][EM_B];
  const int m0 = blockIdx.x * 16;
  const int w = threadIdx.x >> 5, lane = threadIdx.x & 31;
  const int col = lane & 15, rofs = (lane >> 4) << 3;

  // logits (16 x 1024): h_out rows are h_r[:, :128] (ld = 256).
  {
    const int nbase = w * 128;
    v8f acc[8] = {V8ZERO, V8ZERO, V8ZERO, V8ZERO, V8ZERO, V8ZERO, V8ZERO, V8ZERO};
    for (int k0 = 0; k0 < EM_D; k0 += 4) {
      v2f a = load_a(h_r, EM_2D, m0, k0);
      #pragma unroll
      for (int t = 0; t < 8; ++t)
        acc[t] = wmma4(a, load_bT(sg, EM_D, nbase + t * 16, k0), acc[t]);
    }
    #pragma unroll
    for (int t = 0; t < 8; ++t)
      #pragma unroll
      for (int r = 0; r < 8; ++r)
        s_l[rofs + r][nbase + t * 16 + col] = acc[t][r];
  }
  __syncthreads();

  // row softmax: 2 rows per wave, wave32 shuffle reduce.
  for (int i = 0; i < 2; ++i) {
    int row = w * 2 + i;
    float mx = -3.0e38f;
    for (int j = lane; j < EM_B; j += 32) mx = fmaxf(mx, s_l[row][j]);
    for (int off = 16; off > 0; off >>= 1) mx = fmaxf(mx, __shfl_xor(mx, off, 32));
    float sum = 0.f;
    for (int j = lane; j < EM_B; j += 32) {
      float e = expf(s_l[row][j] - mx);
      s_l[row][j] = e; sum += e;
    }
    for (int off = 16; off > 0; off >>= 1) sum += __shfl_xor(sum, off, 32);
    float inv = 1.0f / sum;
    for (int j = lane; j < EM_B; j += 32) s_l[row][j] *= inv;
  }
  __syncthreads();

  // r = attn(16x1024) @ sg(1024x128): one 16-col tile per wave, K = 1024.
  {
    int n0 = w * 16;
    v8f acc = V8ZERO;
    for (int k0 = 0; k0 < EM_B; k0 += 4)
      acc = wmma4(load_a(&s_l[0][0], EM_B, 0, k0), load_b(sg, EM_D, k0, n0), acc);
    #pragma unroll
    for (int r = 0; r < 8; ++r)
      h_r[(long)(m0 + rofs + r) * EM_2D + EM_D + n0 + col] = acc[r];
  }
}

// ---------------------------------------------------------------------------
// K8: cosine similarity per row: out = dot / sqrt(|h|^2 * |s|^2).
// ---------------------------------------------------------------------------
__global__ void k_cos(const float* __restrict__ h_r, const float* __restrict__ sg,
                      float* __restrict__ out) {
  int row = blockIdx.x, lane = threadIdx.x;
  const float* hq = h_r + (long)row * EM_2D;  // h_out = first 128 cols
  const float* sv = sg + (long)row * EM_D;
  float cr = 0.f, n1 = 0.f, n2 = 0.f;
  #pragma unroll
  for (int j = 0; j < 4; ++j) {
    float a = hq[lane * 4 + j], b = sv[lane * 4 + j];
    cr += a * b; n1 += a * a; n2 += b * b;
  }
  for (int off = 16; off > 0; off >>= 1) {
    cr += __shfl_xor(cr, off, 32);
    n1 += __shfl_xor(n1, off, 32);
    n2 += __shfl_xor(n2, off, 32);
  }
  if (lane == 0) out[row] = cr / sqrtf(n1 * n2);
}

// ---------------------------------------------------------------------------
extern "C" void kernel_launch(void* const* d_in, const int* in_sizes, int n_in,
                              void* d_out, int out_size, void* d_ws, size_t ws_size,
                              hipStream_t stream) {
  (void)in_sizes; (void)n_in; (void)out_size; (void)ws_size;
  const int*   rel   = (const int*)d_in[0];
  const int*   ent   = (const int*)d_in[1];
  const int*   query = (const int*)d_in[2];
  const float* emb   = (const float*)d_in[3];
  const float* gcn_w = (const float*)d_in[4];
  const float* gcn_b = (const float*)d_in[5];
  const float* p1w   = (const float*)d_in[6];
  const float* p1b   = (const float*)d_in[7];
  const float* p2w   = (const float*)d_in[8];
  const float* p2b   = (const float*)d_in[9];
  const float* ln_a  = (const float*)d_in[10];
  const float* ln_b  = (const float*)d_in[11];
  const float* w_ih  = (const float*)d_in[12];
  const float* w_hh  = (const float*)d_in[13];
  const float* b_ih  = (const float*)d_in[14];
  const float* b_hh  = (const float*)d_in[15];

  float* ws    = (float*)d_ws;            // ~12.6 MB total
  float* S     = ws;                      // B*256
  float* sg    = S     + EM_B * EM_2D;    // B*128
  float* q_emb = sg    + EM_B * EM_D;     // B*128
  float* qW    = q_emb + EM_B * EM_D;     // B*1024
  float* h_r   = qW    + EM_B * EM_8D;    // B*256
  float* cst   = h_r   + EM_B * EM_2D;    // B*256
  float* gates = cst   + EM_B * EM_2D;    // B*1024

  k_gather_sum<<<EM_B, 256, 0, stream>>>(rel, ent, emb, S);
  k_support<<<EM_B / 16, 128, 0, stream>>>(S, gcn_w, gcn_b, p1w, p1b, p2w, p2b, ln_a, ln_b, sg);
  k_qemb<<<EM_B, 32, 0, stream>>>(query, emb, q_emb);
  k_qw<<<EM_B / 16, 256, 0, stream>>>(query, emb, w_ih, b_ih, b_hh, qW);
  k_zero2<<<EM_B, 256, 0, stream>>>(h_r, cst);
  for (int step = 0; step < 4; ++step) {
    k_gates<<<EM_B / 16, 256, 0, stream>>>(h_r, w_hh, qW, gates);
    k_lstm<<<EM_B, 256, 0, stream>>>(gates, q_emb, cst, h_r);
    k_attn<<<EM_B / 16, 256, 0, stream>>>(h_r, sg);
  }
  k_cos<<<EM_B, 32, 0, stream>>>(h_r, sg, (float*)d_out);
}